// GCNModel_5153960755350
// MI455X (gfx1250) — compile-verified
//
#include <hip/hip_runtime.h>
#include <hip/hip_bf16.h>

typedef __attribute__((ext_vector_type(2))) float f32x2;
typedef __attribute__((ext_vector_type(8))) float f32x8;

#define N_FEAT1 64
#define N_FEAT2 128

// ---------------------------------------------------------------- utilities
__global__ __launch_bounds__(256) void k_fill(float* __restrict__ p,
                                              long long n, float v) {
    long long i = (long long)blockIdx.x * 256 + threadIdx.x;
    if (i < n) p[i] = v;
}

// deg[dst] += 1 for every real edge (deg pre-initialized to 1.0 == self loop)
__global__ __launch_bounds__(256) void k_degree(const int* __restrict__ dst,
                                                float* __restrict__ deg, int E) {
    int e = blockIdx.x * 256 + threadIdx.x;
    if (e < E) atomicAdd(&deg[dst[e]], 1.0f);
}

__global__ __launch_bounds__(256) void k_dinv(const float* __restrict__ deg,
                                              float* __restrict__ dinv, int n) {
    int v = blockIdx.x * 256 + threadIdx.x;
    if (v < n) {
        float d = deg[v];
        dinv[v] = d > 0.0f ? rsqrtf(d) : 0.0f;
    }
}

// ------------------------------------------------- layer 1 (aggregate dim 1)
__global__ __launch_bounds__(256) void k_agg_edges_1(
    const int* __restrict__ src, const int* __restrict__ dst,
    const float* __restrict__ dinv, const float* __restrict__ x,
    float* __restrict__ agg, int E) {
    int e = blockIdx.x * 256 + threadIdx.x;
    if (e < E) {
        int s = src[e], d = dst[e];
        atomicAdd(&agg[d], dinv[s] * dinv[d] * x[s]);
    }
}

// self-loop + rank-1 expansion: h1[v][j] = relu((agg[v] + dinv^2 x[v]) * W1[j] + b1[j])
__global__ __launch_bounds__(256) void k_layer1_node(
    const float* __restrict__ agg, const float* __restrict__ x,
    const float* __restrict__ dinv, const float* __restrict__ W1,
    const float* __restrict__ b1, float* __restrict__ h1, int n) {
    long long t = (long long)blockIdx.x * 256 + threadIdx.x;
    if (t >= (long long)n * N_FEAT1) return;
    int v = (int)(t >> 6);
    int j = (int)(t & (N_FEAT1 - 1));
    float di = dinv[v];
    float a = agg[v] + di * di * x[v];
    float h = a * W1[j] + b1[j];
    h1[t] = h > 0.0f ? h : 0.0f;
}

// ------------------------------------------- layer 2 aggregation (64 floats)
__global__ __launch_bounds__(256) void k_agg_edges_64(
    const int* __restrict__ src, const int* __restrict__ dst,
    const float* __restrict__ dinv, const float* __restrict__ h1,
    float* __restrict__ agg, int E) {
    long long t = (long long)blockIdx.x * 256 + threadIdx.x;
    if (t >= (long long)E * N_FEAT1) return;
    int e = (int)(t >> 6);
    int j = (int)(t & (N_FEAT1 - 1));
    int s = src[e], d = dst[e];
    float nrm = dinv[s] * dinv[d];
    atomicAdd(&agg[(long long)d * N_FEAT1 + j], nrm * h1[(long long)s * N_FEAT1 + j]);
}

__global__ __launch_bounds__(256) void k_selfloop_64(
    const float* __restrict__ dinv, const float* __restrict__ h1,
    float* __restrict__ agg, int n) {
    long long t = (long long)blockIdx.x * 256 + threadIdx.x;
    if (t >= (long long)n * N_FEAT1) return;
    int v = (int)(t >> 6);
    float di = dinv[v];
    agg[t] += di * di * h1[t];
}

// ------------------------- layer 2 GEMM: h2 = relu(agg2[M,64] @ W2[64,128] + b2)
// One wave32 computes one 16x16 tile with V_WMMA_F32_16X16X4_F32.
// 8 waves per block cover N=128; blockIdx.x covers 16 rows of M.
__global__ __launch_bounds__(256) void k_gemm_relu_wmma(
    const float* __restrict__ A,     // [M, 64]
    const float* __restrict__ W,     // [64, 128]
    const float* __restrict__ bias,  // [128]
    float* __restrict__ H,           // [M, 128]
    int M) {
    const int lane = threadIdx.x & 31;
    const int wave = threadIdx.x >> 5;          // N-tile index (0..7)
    const int m0 = blockIdx.x * 16;
    if (m0 + 16 > M) return;                    // wave-uniform: EXEC stays all-ones
    const int n0 = wave * 16;
    const int half = lane >> 4;                 // 0: lanes 0-15, 1: lanes 16-31
    const int mrow = m0 + (lane & 15);          // A-matrix row owned by this lane
    const int ncol = n0 + (lane & 15);          // B/C column owned by this lane

    f32x8 c = {};
    #pragma unroll
    for (int k0 = 0; k0 < N_FEAT1; k0 += 4) {
        const int ka = k0 + 2 * half;
        // A 16x4 layout: lane holds K={ka, ka+1} of its row -> contiguous b64 load
        f32x2 a;
        a.x = A[(long long)mrow * N_FEAT1 + ka];
        a.y = A[(long long)mrow * N_FEAT1 + ka + 1];
        // B 4x16 layout: VGPR0 = row ka, VGPR1 = row ka+1, col = lane%16
        f32x2 b;
        b.x = W[ka * N_FEAT2 + ncol];
        b.y = W[(ka + 1) * N_FEAT2 + ncol];
        c = __builtin_amdgcn_wmma_f32_16x16x4_f32(
                /*neg_a=*/false, a, /*neg_b=*/false, b,
                /*c_mod=*/(short)0, c, /*reuse_a=*/false, /*reuse_b=*/false);
    }

    const float bv = bias[ncol];
    #pragma unroll
    for (int r = 0; r < 8; ++r) {
        int row = m0 + r + (half << 3);         // VGPR r: M=r (lo half), M=r+8 (hi half)
        float val = c[r] + bv;
        H[(long long)row * N_FEAT2 + ncol] = val > 0.0f ? val : 0.0f;
    }
}

// ------------------------------------------------- layer 3: transform then agg
__global__ __launch_bounds__(256) void k_dot128(
    const float* __restrict__ h2, const float* __restrict__ W3,
    float* __restrict__ t, int n) {
    int v = blockIdx.x * 256 + threadIdx.x;
    if (v >= n) return;
    const float4* __restrict__ hv = (const float4*)(h2 + (long long)v * N_FEAT2);
    const float4* __restrict__ wv = (const float4*)W3;
    float acc = 0.0f;
    #pragma unroll
    for (int i = 0; i < N_FEAT2 / 4; ++i) {
        float4 h = hv[i];
        float4 w = wv[i];
        acc = fmaf(h.x, w.x, acc);
        acc = fmaf(h.y, w.y, acc);
        acc = fmaf(h.z, w.z, acc);
        acc = fmaf(h.w, w.w, acc);
    }
    t[v] = acc;
}

// out[v] = dinv^2 * t[v] + b3  (self-loop + bias init, before edge atomics)
__global__ __launch_bounds__(256) void k_layer3_init(
    const float* __restrict__ dinv, const float* __restrict__ t,
    const float* __restrict__ b3, float* __restrict__ out, int n) {
    int v = blockIdx.x * 256 + threadIdx.x;
    if (v < n) {
        float di = dinv[v];
        out[v] = di * di * t[v] + b3[0];
    }
}

__global__ __launch_bounds__(256) void k_agg_edges_out(
    const int* __restrict__ src, const int* __restrict__ dst,
    const float* __restrict__ dinv, const float* __restrict__ t,
    float* __restrict__ out, int E) {
    int e = blockIdx.x * 256 + threadIdx.x;
    if (e < E) {
        int s = src[e], d = dst[e];
        atomicAdd(&out[d], dinv[s] * dinv[d] * t[s]);
    }
}

// ---------------------------------------------------------------- launcher
extern "C" void kernel_launch(void* const* d_in, const int* in_sizes, int n_in,
                              void* d_out, int out_size, void* d_ws, size_t ws_size,
                              hipStream_t stream) {
    const float* x   = (const float*)d_in[0];
    const int*   ei  = (const int*)d_in[1];
    const float* W1  = (const float*)d_in[2];
    const float* b1  = (const float*)d_in[3];
    const float* W2  = (const float*)d_in[4];
    const float* b2  = (const float*)d_in[5];
    const float* W3  = (const float*)d_in[6];
    const float* b3  = (const float*)d_in[7];
    float* out = (float*)d_out;

    const int n = in_sizes[0];          // 100000 nodes
    const int E = in_sizes[1] / 2;      // 1.6M edges
    const int* src = ei;
    const int* dst = ei + E;

    // workspace layout (floats)
    float* ws   = (float*)d_ws;
    float* dinv = ws;                                   // [n]
    float* aggx = dinv + n;                             // [n]
    float* tbuf = aggx + n;                             // [n]
    float* deg  = tbuf + n;                             // [n]
    float* h1   = deg  + n;                             // [n,64]
    float* agg2 = h1   + (long long)n * N_FEAT1;        // [n,64]
    float* h2   = agg2 + (long long)n * N_FEAT1;        // [n,128]

    const int TB = 256;
    auto blocks = [](long long work) { return (unsigned)((work + 255) / 256); };

    // degrees (deg starts at 1.0 == self loop) and D^{-1/2}
    k_fill<<<blocks(n), TB, 0, stream>>>(deg, n, 1.0f);
    k_fill<<<blocks(n), TB, 0, stream>>>(aggx, n, 0.0f);
    k_fill<<<blocks((long long)n * N_FEAT1), TB, 0, stream>>>(agg2, (long long)n * N_FEAT1, 0.0f);
    k_degree<<<blocks(E), TB, 0, stream>>>(dst, deg, E);
    k_dinv<<<blocks(n), TB, 0, stream>>>(deg, dinv, n);

    // layer 1: aggregate scalar x, then rank-1 expand to 64 feats + ReLU
    k_agg_edges_1<<<blocks(E), TB, 0, stream>>>(src, dst, dinv, x, aggx, E);
    k_layer1_node<<<blocks((long long)n * N_FEAT1), TB, 0, stream>>>(
        aggx, x, dinv, W1, b1, h1, n);

    // layer 2: aggregate 64-wide, then WMMA GEMM (64->128) + bias + ReLU
    k_agg_edges_64<<<blocks((long long)E * N_FEAT1), TB, 0, stream>>>(
        src, dst, dinv, h1, agg2, E);
    k_selfloop_64<<<blocks((long long)n * N_FEAT1), TB, 0, stream>>>(
        dinv, h1, agg2, n);
    k_gemm_relu_wmma<<<(n + 15) / 16, TB, 0, stream>>>(agg2, W2, b2, h2, n);

    // layer 3: transform 128->1 first, then aggregate scalar
    k_dot128<<<blocks(n), TB, 0, stream>>>(h2, W3, tbuf, n);
    k_layer3_init<<<blocks(n), TB, 0, stream>>>(dinv, tbuf, b3, out, n);
    k_agg_edges_out<<<blocks(E), TB, 0, stream>>>(src, dst, dinv, tbuf, out, E);
}